// MultichannelSequentialLinearLayerWithIndependentAndParallelChannels_82059645157739
// MI455X (gfx1250) — compile-verified
//
#include <hip/hip_runtime.h>
#include <stdint.h>

typedef __attribute__((ext_vector_type(2))) float v2f;
typedef __attribute__((ext_vector_type(8))) float v8f;

#define IC   8
#define RC   8
#define PC   8
#define G    4
#define IPG  128
#define OPG  128
#define ROWS_PER_B   (RC * PC * G)        // 256 rows per (batch, channel)
#define ROW_STRIDE_B (IC * ROWS_PER_B)    // 2048 global rows per batch index
#define M_TILE 64
#define LDS_STRIDE (IPG + 4)              // 132 floats: conflict-free A reads

__global__ __launch_bounds__(256)
void grouped_linear_wmma_f32(const float* __restrict__ x,
                             const float* __restrict__ W,
                             const float* __restrict__ bias,
                             float* __restrict__ y)
{
    __shared__ float ldsA[M_TILE * LDS_STRIDE];   // 33792 B of 320 KB

    const int ch    = blockIdx.y;                 // independent channel i
    const int tile  = blockIdx.x;                 // M tile within channel
    const int rowc0 = tile * M_TILE;              // channel-local row base
    const int bidx  = rowc0 >> 8;                 // which batch b (256 rows per b)
    const int rpg0  = rowc0 & 255;                // (r,p,g) row within the b-block
    const long gRowBase = (long)bidx * ROW_STRIDE_B + (long)ch * ROWS_PER_B + rpg0;

    const int tid    = threadIdx.x;
    const int lane   = tid & 31;
    const int wave   = tid >> 5;
    const int lane15 = lane & 15;
    const int hi     = (lane >> 4) & 1;           // lanes 16-31 handle K+2 / M+8 halves
    const int n0     = wave * 16;                 // this wave's 16 output columns

    // ---- Stage A tile (64 x 128 f32) global -> LDS via async DMA path ----
    // GLOBAL_LOAD_ASYNC_TO_LDS_B128: no VGPR round-trip, tracked by ASYNCcnt.
    const float* xtile = x + gRowBase * IPG;      // uniform -> SGPR pair (saddr)
    #pragma unroll
    for (int it = 0; it < (M_TILE * IPG) / (4 * 256); ++it) {
        const int idx  = it * 256 + tid;          // float4 index within tile
        const int e    = idx * 4;
        const int row  = e >> 7;                  // / 128
        const int col  = e & 127;
        const unsigned lds_off = (unsigned)(uintptr_t)&ldsA[row * LDS_STRIDE + col];
        const int      voff    = idx * 16;        // per-lane global byte offset
        asm volatile("global_load_async_to_lds_b128 %0, %1, %2"
                     :
                     : "v"(lds_off), "v"(voff), "s"(xtile)
                     : "memory");
    }

    // ---- Preload B fragments: W[ch][n0+lane15][*], per-lane contiguous K pairs ----
    // (overlaps with the async LDS fill; separate LOADcnt domain)
    // B(4x16) layout: VGPR0 = B[k0+2*hi][n], VGPR1 = B[k0+2*hi+1][n]
    const float* wrow = W + ((long)ch * OPG + n0 + lane15) * IPG + 2 * hi;
    v2f bfrag[32];
    #pragma unroll
    for (int ks = 0; ks < 32; ++ks)
        bfrag[ks] = *(const v2f*)(wrow + 4 * ks);

    // Fence the async tile copy, then rendezvous all 8 waves.
    asm volatile("s_wait_asynccnt 0x0" ::: "memory");
    __syncthreads();

    // ---- K loop: 32 steps of V_WMMA_F32_16X16X4_F32 per 16x16 sub-tile ----
    v8f c0 = {}, c1 = {}, c2 = {}, c3 = {};
    #pragma unroll
    for (int ks = 0; ks < 32; ++ks) {
        const int k0 = 4 * ks + 2 * hi;           // A: lanes0-15 -> K=k0, lanes16-31 -> K=k0+2
        const v2f a0 = *(const v2f*)&ldsA[( 0 + lane15) * LDS_STRIDE + k0];
        const v2f a1 = *(const v2f*)&ldsA[(16 + lane15) * LDS_STRIDE + k0];
        const v2f a2 = *(const v2f*)&ldsA[(32 + lane15) * LDS_STRIDE + k0];
        const v2f a3 = *(const v2f*)&ldsA[(48 + lane15) * LDS_STRIDE + k0];
        c0 = __builtin_amdgcn_wmma_f32_16x16x4_f32(false, a0, false, bfrag[ks], (short)0, c0, false, false);
        c1 = __builtin_amdgcn_wmma_f32_16x16x4_f32(false, a1, false, bfrag[ks], (short)0, c1, false, false);
        c2 = __builtin_amdgcn_wmma_f32_16x16x4_f32(false, a2, false, bfrag[ks], (short)0, c2, false, false);
        c3 = __builtin_amdgcn_wmma_f32_16x16x4_f32(false, a3, false, bfrag[ks], (short)0, c3, false, false);
    }

    // ---- Epilogue: bias (per output column) + store ----
    const float bv = bias[ch * OPG + n0 + lane15];
    float* og = y + gRowBase * OPG + n0 + lane15;
    #pragma unroll
    for (int v = 0; v < 8; ++v) {
        const int r = v + 8 * hi;                 // C/D layout: VGPR v -> rows v / v+8
        og[(long)( 0 + r) * OPG] = c0[v] + bv;
        og[(long)(16 + r) * OPG] = c1[v] + bv;
        og[(long)(32 + r) * OPG] = c2[v] + bv;
        og[(long)(48 + r) * OPG] = c3[v] + bv;
    }
}

extern "C" void kernel_launch(void* const* d_in, const int* in_sizes, int n_in,
                              void* d_out, int out_size, void* d_ws, size_t ws_size,
                              hipStream_t stream) {
    (void)in_sizes; (void)n_in; (void)out_size; (void)d_ws; (void)ws_size;
    const float* x = (const float*)d_in[0];   // (B, IC, RC, PC, G, IPG) contiguous
    const float* W = (const float*)d_in[1];   // (IC, OPG, IPG)
    const float* b = (const float*)d_in[2];   // (IC, OPG)
    float* out = (float*)d_out;               // (B, IC, RC, PC, G, OPG)

    const int B = 128;
    const int m_tiles = (B * ROWS_PER_B) / M_TILE;   // 32768 / 64 = 512
    dim3 grid(m_tiles, IC);
    dim3 block(256);
    grouped_linear_wmma_f32<<<grid, block, 0, stream>>>(x, W, b, out);
}